// Dgcn_85452669321742
// MI455X (gfx1250) — compile-verified
//
#include <hip/hip_runtime.h>
#include <hip/hip_bf16.h>

// Problem constants (from reference): B=32, N=512, D=768, M=K+1=4
#define B_SZ   32
#define N_SZ   512
#define D_SZ   768
#define M_SZ   4
#define CDB    (D_SZ * B_SZ)      // 24576  (columns of diffusion GEMMs)
#define OUTC   D_SZ               // 768    (output columns)
// Workspace layout (floats): support[512*512] | dinv[512] | x0..x3[512*24576 each]
// Total = (262144 + 512 + 4*12582912) * 4 bytes ~= 193 MiB

typedef __attribute__((ext_vector_type(2))) float v2f;
typedef __attribute__((ext_vector_type(8))) float v8f;

__device__ __forceinline__ v8f wmma4(v2f a, v2f b, v8f c) {
  // V_WMMA_F32_16X16X4_F32 : D = A(16x4 f32) * B(4x16 f32) + C(16x16 f32)
  return __builtin_amdgcn_wmma_f32_16x16x4_f32(
      /*neg_a=*/false, a, /*neg_b=*/false, b,
      /*c_mod=*/(short)0, c, /*reuse_a=*/false, /*reuse_b=*/false);
}

// ---------------------------------------------------------------------------
// Kernel 1: per-row degree -> d_inv_sqrt.  rowsum(adj + I) = sum(adj_row) + 1
// ---------------------------------------------------------------------------
__global__ __launch_bounds__(256)
void dinv_kernel(const float* __restrict__ adj, float* __restrict__ dinv) {
  __shared__ float red[256];
  const int row = blockIdx.x;
  const int t = threadIdx.x;
  float s = 0.f;
  const float* a = adj + row * N_SZ;
  for (int j = t; j < N_SZ; j += 256) s += a[j];
  red[t] = s;
  __syncthreads();
  for (int off = 128; off > 0; off >>= 1) {
    if (t < off) red[t] += red[t + off];
    __syncthreads();
  }
  if (t == 0) {
    float rs = red[0] + 1.0f;  // + self loop on the diagonal
    dinv[row] = (rs > 0.f) ? rsqrtf(rs) : 0.f;
  }
}

// ---------------------------------------------------------------------------
// Kernel 2: support[i,j] = (adj[j,i] + (i==j)) * d[i] * d[j]
//           ( == ((m * d[None,:]).T * d[None,:]) with m = adj + I )
// ---------------------------------------------------------------------------
__global__ __launch_bounds__(256)
void support_kernel(const float* __restrict__ adj, const float* __restrict__ dinv,
                    float* __restrict__ support) {
  const int idx = blockIdx.x * 256 + threadIdx.x;   // 512*512 total
  const int i = idx >> 9;
  const int j = idx & (N_SZ - 1);
  float m = adj[j * N_SZ + i] + (i == j ? 1.0f : 0.0f);
  support[idx] = m * dinv[i] * dinv[j];
}

// ---------------------------------------------------------------------------
// Kernel 3: x0[n, d*B + b] = inputs[b, n, d]   (LDS-tiled transpose)
// grid = (N, D/64); block = 256. Tile: 64 (d) x 32 (b).
// ---------------------------------------------------------------------------
__global__ __launch_bounds__(256)
void transpose_kernel(const float* __restrict__ inputs, float* __restrict__ x0) {
  __shared__ float tile[64 * 33];  // padded: odd stride -> no bank conflicts
  const int n  = blockIdx.x;
  const int d0 = blockIdx.y * 64;
  const int t  = threadIdx.x;
  // Load: coalesced along d for each b
#pragma unroll
  for (int p = 0; p < 8; ++p) {
    int b  = p * 4 + (t >> 6);
    int dl = t & 63;
    tile[dl * 33 + b] =
        inputs[(size_t)b * (N_SZ * D_SZ) + (size_t)n * D_SZ + d0 + dl];
  }
  __syncthreads();
  // Store: coalesced along the packed (d*32 + b) column index
  float* dst = x0 + (size_t)n * CDB + d0 * B_SZ;
#pragma unroll
  for (int p = 0; p < 8; ++p) {
    int c  = p * 256 + t;
    int dl = c >> 5;
    int b  = c & 31;
    dst[c] = tile[dl * 33 + b];
  }
}

// ---------------------------------------------------------------------------
// Epilogue helper: write one 16x64 C tile with Y = alpha*acc + beta*Xprev
// ---------------------------------------------------------------------------
__device__ __forceinline__
void store_tile(float* __restrict__ Y, const float* __restrict__ Xprev,
                int rbase, int col, float alpha, float beta,
                v8f a0, v8f a1, v8f a2, v8f a3) {
  if (beta != 0.0f) {
#pragma unroll
    for (int j = 0; j < 8; ++j) {
      float* Yr = Y + (rbase + j) * CDB + col;
      const float* Pr = Xprev + (rbase + j) * CDB + col;
      Yr[0]  = alpha * a0[j] + beta * Pr[0];
      Yr[16] = alpha * a1[j] + beta * Pr[16];
      Yr[32] = alpha * a2[j] + beta * Pr[32];
      Yr[48] = alpha * a3[j] + beta * Pr[48];
    }
  } else {
#pragma unroll
    for (int j = 0; j < 8; ++j) {
      float* Yr = Y + (rbase + j) * CDB + col;
      Yr[0]  = alpha * a0[j];
      Yr[16] = alpha * a1[j];
      Yr[32] = alpha * a2[j];
      Yr[48] = alpha * a3[j];
    }
  }
}

// ---------------------------------------------------------------------------
// Kernel 4: Y = alpha * (S @ X) + beta * Xprev          (S: 512x512, X: 512xCDB)
// Wave tile: 32(M) x 64(N) -> 8 accumulators; B fragments reused across 2 M rows
// (10 vmem ops per 8 WMMAs). WG = 8 waves as 2(M) x 4(N) => 64 x 256 of C.
// grid = (CDB/256, N/64) = (96, 8)
// A-frag layout (16x4 f32): lane&15 = M row; lane>>4 selects K pair {0,1}/{2,3}.
// B-frag layout (4x16 f32): lane&15 = N col; VGPR0/1 = K rows (2*kh)/(2*kh+1).
// ---------------------------------------------------------------------------
__global__ __launch_bounds__(256)
void diffusion_gemm(const float* __restrict__ S, const float* __restrict__ X,
                    const float* __restrict__ Xprev, float* __restrict__ Y,
                    float alpha, float beta) {
  const int lane = threadIdx.x & 31;
  const int wave = threadIdx.x >> 5;
  const int lm   = lane & 15;
  const int kh   = lane >> 4;
  const int m0 = blockIdx.y * 64 + (wave >> 2) * 32;
  const int n0 = blockIdx.x * 256 + (wave & 3) * 64;

  const float* Srow0 = S + (m0 + lm) * N_SZ + kh * 2;  // rows m0..m0+15
  const float* Srow1 = Srow0 + 16 * N_SZ;              // rows m0+16..m0+31
  const float* Xcol  = X + n0 + lm;

  v8f acc00 = {}, acc01 = {}, acc02 = {}, acc03 = {};
  v8f acc10 = {}, acc11 = {}, acc12 = {}, acc13 = {};
  for (int k = 0; k < N_SZ; k += 4) {
    v2f a0 = *(const v2f*)(Srow0 + k);
    v2f a1 = *(const v2f*)(Srow1 + k);
    const float* Xr0 = Xcol + (k + kh * 2) * CDB;
    const float* Xr1 = Xr0 + CDB;
    v2f b0 = {Xr0[0],  Xr1[0]};
    v2f b1 = {Xr0[16], Xr1[16]};
    v2f b2 = {Xr0[32], Xr1[32]};
    v2f b3 = {Xr0[48], Xr1[48]};
    acc00 = wmma4(a0, b0, acc00);
    acc01 = wmma4(a0, b1, acc01);
    acc02 = wmma4(a0, b2, acc02);
    acc03 = wmma4(a0, b3, acc03);
    acc10 = wmma4(a1, b0, acc10);
    acc11 = wmma4(a1, b1, acc11);
    acc12 = wmma4(a1, b2, acc12);
    acc13 = wmma4(a1, b3, acc13);
  }

  // C layout: VGPR j -> row rbase + j with rbase = tile_row + 8*kh
  store_tile(Y, Xprev, m0 + kh * 8,      n0 + lm, alpha, beta,
             acc00, acc01, acc02, acc03);
  store_tile(Y, Xprev, m0 + 16 + kh * 8, n0 + lm, alpha, beta,
             acc10, acc11, acc12, acc13);
}

// ---------------------------------------------------------------------------
// Kernel 5: out[b*N+n, o] = sum_{d,m} x_m[n, d*B+b] * W[d*4+m, o] + bias[o]
// Each K=4 WMMA step consumes one d across all four diffusion terms:
//   lane-half kh=0 reads (x0, x1), kh=1 reads (x2, x3).
// Wave tile 32(M) x 64(N); grid = (OUTC/256, (B*N)/64) = (3, 256)
// ---------------------------------------------------------------------------
__global__ __launch_bounds__(256)
void output_gemm(const float* __restrict__ x0, const float* __restrict__ x1,
                 const float* __restrict__ x2, const float* __restrict__ x3,
                 const float* __restrict__ W, const float* __restrict__ bias,
                 float* __restrict__ out) {
  const int lane = threadIdx.x & 31;
  const int wave = threadIdx.x >> 5;
  const int lm   = lane & 15;
  const int kh   = lane >> 4;
  const int r0 = blockIdx.y * 64 + (wave >> 2) * 32;
  const int o0 = blockIdx.x * 256 + (wave & 3) * 64;

  // A row for this lane, both 16-row halves of the 32-row tile
  const int ra = r0 + lm;
  const int rb = r0 + 16 + lm;
  const size_t aoff0 = (size_t)(ra & (N_SZ - 1)) * CDB + (ra >> 9);
  const size_t aoff1 = (size_t)(rb & (N_SZ - 1)) * CDB + (rb >> 9);
  const float* xa = (kh == 0) ? x0 : x2;  // K offset 2*kh + 0
  const float* xb = (kh == 0) ? x1 : x3;  // K offset 2*kh + 1
  const float* Wb = W + kh * 2 * OUTC + o0 + lm;

  v8f acc00 = {}, acc01 = {}, acc02 = {}, acc03 = {};
  v8f acc10 = {}, acc11 = {}, acc12 = {}, acc13 = {};
  for (int d = 0; d < D_SZ; ++d) {
    const int xo = d * B_SZ;
    v2f a0 = {xa[aoff0 + xo], xb[aoff0 + xo]};
    v2f a1 = {xa[aoff1 + xo], xb[aoff1 + xo]};
    const float* Wr0 = Wb + (d * 4) * OUTC;
    const float* Wr1 = Wr0 + OUTC;
    v2f b0 = {Wr0[0],  Wr1[0]};
    v2f b1 = {Wr0[16], Wr1[16]};
    v2f b2 = {Wr0[32], Wr1[32]};
    v2f b3 = {Wr0[48], Wr1[48]};
    acc00 = wmma4(a0, b0, acc00);
    acc01 = wmma4(a0, b1, acc01);
    acc02 = wmma4(a0, b2, acc02);
    acc03 = wmma4(a0, b3, acc03);
    acc10 = wmma4(a1, b0, acc10);
    acc11 = wmma4(a1, b1, acc11);
    acc12 = wmma4(a1, b2, acc12);
    acc13 = wmma4(a1, b3, acc13);
  }

  const float bi0 = bias[o0 + lm];
  const float bi1 = bias[o0 + 16 + lm];
  const float bi2 = bias[o0 + 32 + lm];
  const float bi3 = bias[o0 + 48 + lm];
#pragma unroll
  for (int h = 0; h < 2; ++h) {
    const int rbase = r0 + h * 16 + kh * 8;
    v8f c0 = h ? acc10 : acc00;
    v8f c1 = h ? acc11 : acc01;
    v8f c2 = h ? acc12 : acc02;
    v8f c3 = h ? acc13 : acc03;
#pragma unroll
    for (int j = 0; j < 8; ++j) {
      float* Or = out + (rbase + j) * OUTC + o0 + lm;
      Or[0]  = c0[j] + bi0;
      Or[16] = c1[j] + bi1;
      Or[32] = c2[j] + bi2;
      Or[48] = c3[j] + bi3;
    }
  }
}

// ---------------------------------------------------------------------------
extern "C" void kernel_launch(void* const* d_in, const int* in_sizes, int n_in,
                              void* d_out, int out_size, void* d_ws, size_t ws_size,
                              hipStream_t stream) {
  const float* inputs  = (const float*)d_in[0];  // [32, 512, 768]
  const float* adj     = (const float*)d_in[1];  // [512, 512]
  const float* weights = (const float*)d_in[2];  // [3072, 768]
  const float* biases  = (const float*)d_in[3];  // [768]
  float* out = (float*)d_out;                    // [32, 512, 768] flattened

  float* ws      = (float*)d_ws;
  float* support = ws;                                  // 512*512
  float* dinv    = support + (size_t)N_SZ * N_SZ;       // 512
  float* x0      = dinv + 512;                          // each 512*24576
  float* x1      = x0 + (size_t)N_SZ * CDB;
  float* x2      = x1 + (size_t)N_SZ * CDB;
  float* x3      = x2 + (size_t)N_SZ * CDB;

  (void)in_sizes; (void)n_in; (void)out_size; (void)ws_size;

  dinv_kernel<<<N_SZ, 256, 0, stream>>>(adj, dinv);
  support_kernel<<<(N_SZ * N_SZ) / 256, 256, 0, stream>>>(adj, dinv, support);
  transpose_kernel<<<dim3(N_SZ, D_SZ / 64), 256, 0, stream>>>(inputs, x0);

  dim3 gd(CDB / 256, N_SZ / 64);  // (96, 8)
  // x1 = S @ x0
  diffusion_gemm<<<gd, 256, 0, stream>>>(support, x0, x0, x1, 1.0f, 0.0f);
  // x2 = 2*(S @ x1) - x0
  diffusion_gemm<<<gd, 256, 0, stream>>>(support, x1, x0, x2, 2.0f, -1.0f);
  // x3 = 2*(S @ x2) - x1
  diffusion_gemm<<<gd, 256, 0, stream>>>(support, x2, x1, x3, 2.0f, -1.0f);

  // out = gather([x0..x3]) @ W + b
  output_gemm<<<dim3(OUTC / 256, (B_SZ * N_SZ) / 64), 256, 0, stream>>>(
      x0, x1, x2, x3, weights, biases, out);
}